// TokenMixer_6588479832293
// MI455X (gfx1250) — compile-verified
//
#include <hip/hip_runtime.h>
#include <hip/hip_bf16.h>

#define B_    32
#define CIN   512
#define GC_   128
#define H_    56
#define W_    56
#define HW_   3136
#define HEADS_ 4
#define HD_   32

typedef float v2f __attribute__((ext_vector_type(2)));
typedef float v8f __attribute__((ext_vector_type(8)));

__device__ inline v8f zero8() {
  v8f z;
#pragma unroll
  for (int i = 0; i < 8; ++i) z[i] = 0.f;
  return z;
}

// ---------------------------------------------------------------------------
// 1) mem passthrough: out[b][0:128] = x[b][0:128]  (float4 streams)
// ---------------------------------------------------------------------------
__global__ void copy_mem_kernel(const float* __restrict__ x, float* __restrict__ out) {
  const size_t i = (size_t)blockIdx.x * blockDim.x + threadIdx.x;   // float4 idx
  const size_t perB_out = (size_t)GC_ * HW_ / 4;                    // 100352
  const size_t perB_in  = (size_t)CIN * HW_ / 4;                    // 401408
  const size_t b = i / perB_out;
  const size_t r = i - b * perB_out;
  const float4* __restrict__ x4 = (const float4*)x;
  float4* __restrict__ o4 = (float4*)out;
  o4[i] = x4[b * perB_in + r];
}

// ---------------------------------------------------------------------------
// 2/3) depthwise conv, KS x KS, per-(b,c) plane, weights in registers
// feats layout: (B, 384, HW): att [0:128), conv1 [128:256), conv2 [256:384)
// ---------------------------------------------------------------------------
template <int KS, int PAD>
__global__ void dwconv_kernel(const float* __restrict__ x, const float* __restrict__ w,
                              const float* __restrict__ bias, float* __restrict__ feats,
                              int in_ch_off, int out_ch_off) {
  const int bc = blockIdx.x;
  const int b = bc >> 7, c = bc & 127;
  const float* __restrict__ xin = x + ((size_t)b * CIN + in_ch_off + c) * HW_;
  float* __restrict__ o = feats + ((size_t)b * 384 + out_ch_off + c) * HW_;
  float wv[KS * KS];
#pragma unroll
  for (int i = 0; i < KS * KS; ++i) wv[i] = w[c * KS * KS + i];
  const float bv = bias[c];
  for (int n = threadIdx.x; n < HW_; n += blockDim.x) {
    const int h = n / W_, wc = n - h * W_;
    float acc = bv;
#pragma unroll
    for (int kh = 0; kh < KS; ++kh) {
      const int hh = h + kh - PAD;
      if (hh < 0 || hh >= H_) continue;
#pragma unroll
      for (int kw = 0; kw < KS; ++kw) {
        const int ww = wc + kw - PAD;
        if (ww < 0 || ww >= W_) continue;
        acc = fmaf(wv[kh * KS + kw], xin[hh * W_ + ww], acc);
      }
    }
    o[n] = acc;
  }
}

// ---------------------------------------------------------------------------
// 4) qk 1x1-conv GEMM per batch: qk[o][n] = sum_c W[o][c] * feat_att[c][n] + b[o]
//    then elu(v)+1 (applies to both q and k halves), stored to workspace.
//    f32 WMMA 16x16x4. Workgroup: 8 waves, n-block = 64, each wave o-strip = 32.
//    X tile staged to LDS via CDNA5 async-to-LDS path (ASYNCcnt).
// ---------------------------------------------------------------------------
#define QK_NBLK 64
__global__ void __launch_bounds__(256) qk_gemm_kernel(const float* __restrict__ x,
                                                      const float* __restrict__ wqk,
                                                      const float* __restrict__ bqk,
                                                      float* __restrict__ qk) {
  // +4 pad: row stride 272B (17x16) keeps every async B128 LDS target 16B-aligned
  __shared__ float Xs[128][QK_NBLK + 4];
  const int b = blockIdx.x / 49;
  const int nb = blockIdx.x - b * 49;
  const int n0 = nb * QK_NBLK;
  const float* __restrict__ xin = x + ((size_t)b * CIN + GC_) * HW_;   // feat_att

  // cooperative stage via GLOBAL_LOAD_ASYNC_TO_LDS_B128: 128 rows x 64 n,
  // 16B per lane per issue, 8 issues per wave; tracked with ASYNCcnt.
  for (int idx = threadIdx.x; idx < 128 * QK_NBLK / 4; idx += 256) {
    const int row = idx >> 4;
    const int col = (idx & 15) << 2;
    const float* gsrc = xin + (size_t)row * HW_ + n0 + col;
    const unsigned lds = (unsigned)(uintptr_t)(&Xs[row][col]);
    asm volatile("global_load_async_to_lds_b128 %0, %1, off"
                 :: "v"(lds), "v"(gsrc) : "memory");
  }
  asm volatile("s_wait_asynccnt 0x0" ::: "memory");
  __syncthreads();

  const int lane = threadIdx.x & 31;
  const int wave = threadIdx.x >> 5;
  const int lhalf = lane & 15;
  const int khi = (lane >> 4) << 1;       // A/B: lanes 16-31 hold K=2,3
  const int o0 = wave * 32;               // 2 o-tiles per wave

  v8f acc[2][4];
#pragma unroll
  for (int i = 0; i < 2; ++i)
#pragma unroll
    for (int j = 0; j < 4; ++j) acc[i][j] = zero8();

  for (int c0 = 0; c0 < 128; c0 += 4) {
    v2f af[2];
#pragma unroll
    for (int ot = 0; ot < 2; ++ot) {
      const int o = o0 + ot * 16 + lhalf;           // A: M across lanes 0-15 (mirrored)
      af[ot].x = wqk[o * 128 + c0 + khi];
      af[ot].y = wqk[o * 128 + c0 + khi + 1];
    }
    v2f bf[4];
#pragma unroll
    for (int nt = 0; nt < 4; ++nt) {                // B: N across lanes, K in vgpr/half
      bf[nt].x = Xs[c0 + khi][nt * 16 + lhalf];
      bf[nt].y = Xs[c0 + khi + 1][nt * 16 + lhalf];
    }
#pragma unroll
    for (int ot = 0; ot < 2; ++ot)
#pragma unroll
      for (int nt = 0; nt < 4; ++nt)
        acc[ot][nt] = __builtin_amdgcn_wmma_f32_16x16x4_f32(
            false, af[ot], false, bf[nt], (short)0, acc[ot][nt], false, false);
  }

  // epilogue: bias + elu(v)+1, store channel-major (16 contiguous n per half-wave)
#pragma unroll
  for (int ot = 0; ot < 2; ++ot)
#pragma unroll
    for (int nt = 0; nt < 4; ++nt)
#pragma unroll
      for (int r = 0; r < 8; ++r) {
        const int o = o0 + ot * 16 + r + ((lane >> 4) << 3);   // C: M = r (+8 hi half)
        const int n = n0 + nt * 16 + lhalf;                    // C: N across lanes
        float v = acc[ot][nt][r] + bqk[o];
        v = v > 0.f ? v + 1.f : __expf(v);
        qk[((size_t)b * 2 * GC_ + o) * HW_ + n] = v;
      }
}

// ---------------------------------------------------------------------------
// 5) kmean[d] = mean_n k[n][d]   (k = qk channels [128,256))
// ---------------------------------------------------------------------------
__global__ void kmean_kernel(const float* __restrict__ qk, float* __restrict__ kmean) {
  const int bh = blockIdx.x;                 // b*4 + h
  const int b = bh >> 2, h = bh & 3;
  const float* __restrict__ ks = qk + ((size_t)b * 256 + GC_ + h * HD_) * HW_;
  const int t = threadIdx.x;
  const int d = t >> 3, part = t & 7;
  float s = 0.f;
  for (int n = part; n < HW_; n += 8) s += ks[(size_t)d * HW_ + n];
  __shared__ float red[256];
  red[t] = s;
  __syncthreads();
  if (part == 0) {
    float tot = 0.f;
#pragma unroll
    for (int i = 0; i < 8; ++i) tot += red[d * 8 + i];
    kmean[bh * HD_ + d] = tot * (1.0f / HW_);
  }
}

// ---------------------------------------------------------------------------
// 6) kv[d][e] = (1/HW) sum_n k[n][d] * v[n][e]  per (b,h); LDS-chunked outer product
// ---------------------------------------------------------------------------
__global__ void __launch_bounds__(256) kv_kernel(const float* __restrict__ qk,
                                                 const float* __restrict__ x,
                                                 float* __restrict__ kv) {
  const int bh = blockIdx.x;
  const int b = bh >> 2, h = bh & 3;
  const float* __restrict__ ks = qk + ((size_t)b * 256 + GC_ + h * HD_) * HW_;  // k^T [d][n]
  const float* __restrict__ vs = x + ((size_t)b * CIN + GC_ + h * HD_) * HW_;   // v^T [e][n]
  __shared__ float kbuf[32][33];
  __shared__ float vbuf[32][33];
  const int t = threadIdx.x;
  const int d = t >> 3;
  const int e0 = (t & 7) * 4;
  float acc[4] = {0.f, 0.f, 0.f, 0.f};
  const int row = t >> 3;
  const int c4 = (t & 7) * 4;
  for (int nc = 0; nc < HW_; nc += 32) {
    float4 k4 = *(const float4*)(ks + (size_t)row * HW_ + nc + c4);
    kbuf[row][c4] = k4.x; kbuf[row][c4 + 1] = k4.y; kbuf[row][c4 + 2] = k4.z; kbuf[row][c4 + 3] = k4.w;
    float4 v4 = *(const float4*)(vs + (size_t)row * HW_ + nc + c4);
    vbuf[row][c4] = v4.x; vbuf[row][c4 + 1] = v4.y; vbuf[row][c4 + 2] = v4.z; vbuf[row][c4 + 3] = v4.w;
    __syncthreads();
#pragma unroll 8
    for (int nn = 0; nn < 32; ++nn) {
      const float kval = kbuf[d][nn];
      acc[0] = fmaf(kval, vbuf[e0 + 0][nn], acc[0]);
      acc[1] = fmaf(kval, vbuf[e0 + 1][nn], acc[1]);
      acc[2] = fmaf(kval, vbuf[e0 + 2][nn], acc[2]);
      acc[3] = fmaf(kval, vbuf[e0 + 3][nn], acc[3]);
    }
    __syncthreads();
  }
#pragma unroll
  for (int j = 0; j < 4; ++j)
    kv[(size_t)bh * (HD_ * HD_) + d * HD_ + e0 + j] = acc[j] * (1.0f / HW_);
}

// ---------------------------------------------------------------------------
// 7) att[n][e] = z[n] * sum_d q[n][d] * kv[d][e]; z[n] = 1/(q.kmean + 1e-6)
//    f32 WMMA, K=32 (8 chained 16x16x4), LDS-transposed coalesced store.
// ---------------------------------------------------------------------------
__global__ void __launch_bounds__(256) att_kernel(const float* __restrict__ qk,
                                                  const float* __restrict__ kvw,
                                                  const float* __restrict__ kmean,
                                                  float* __restrict__ feats) {
  const int bh = blockIdx.y;
  const int nb = blockIdx.x;
  const int b = bh >> 2, h = bh & 3;
  const int n0 = nb * 64;
  const float* __restrict__ qs = qk + ((size_t)b * 256 + h * HD_) * HW_;  // q^T [d][n]

  __shared__ float kvS[32][33];
  __shared__ float kmS[32];
  __shared__ float zred[4][64];
  __shared__ float zS[64];
  __shared__ float oS[32][65];      // [e][n_local]

  const int t = threadIdx.x;
  if (t < 32) kmS[t] = kmean[bh * HD_ + t];
  for (int i = t; i < 1024; i += 256) kvS[i >> 5][i & 31] = kvw[(size_t)bh * 1024 + i];
  __syncthreads();

  // z: 64 n's, 4 d-groups of 8
  {
    const int nl = t & 63, dg = t >> 6;
    float zp = 0.f;
#pragma unroll
    for (int dd = 0; dd < 8; ++dd) {
      const int d = dg * 8 + dd;
      zp = fmaf(qs[(size_t)d * HW_ + n0 + nl], kmS[d], zp);
    }
    zred[dg][nl] = zp;
  }
  __syncthreads();
  if (t < 64) zS[t] = 1.0f / (zred[0][t] + zred[1][t] + zred[2][t] + zred[3][t] + 1e-6f);
  __syncthreads();

  const int lane = t & 31, wave = t >> 5;
  const int lhalf = lane & 15;
  const int khi = (lane >> 4) << 1;
  const int nt = wave & 3, et = wave >> 2;    // 4 n-tiles x 2 e-tiles
  const int nbase = n0 + nt * 16;

  v8f acc = zero8();
#pragma unroll
  for (int d0 = 0; d0 < 32; d0 += 4) {
    v2f a, bb;
    a.x = qs[(size_t)(d0 + khi) * HW_ + nbase + lhalf];       // A[m=n][k=d], coalesced
    a.y = qs[(size_t)(d0 + khi + 1) * HW_ + nbase + lhalf];
    bb.x = kvS[d0 + khi][et * 16 + lhalf];                    // B[k=d][n=e]
    bb.y = kvS[d0 + khi + 1][et * 16 + lhalf];
    acc = __builtin_amdgcn_wmma_f32_16x16x4_f32(false, a, false, bb, (short)0, acc,
                                                false, false);
  }

  // scale by z, transpose through LDS
#pragma unroll
  for (int r = 0; r < 8; ++r) {
    const int nr = nt * 16 + r + ((lane >> 4) << 3);  // local n (0..63)
    const int e = et * 16 + lhalf;
    oS[e][nr] = acc[r] * zS[nr];
  }
  __syncthreads();

  // coalesced channel-major store: feat_att_out occupies feats channels [0,128)
  float* __restrict__ oa = feats + ((size_t)b * 384 + h * HD_) * HW_;
  for (int i = t; i < 32 * 64; i += 256) {
    const int e = i >> 6, nl2 = i & 63;
    oa[(size_t)e * HW_ + n0 + nl2] = oS[e][nl2];
  }
}

// ---------------------------------------------------------------------------
extern "C" void kernel_launch(void* const* d_in, const int* in_sizes, int n_in,
                              void* d_out, int out_size, void* d_ws, size_t ws_size,
                              hipStream_t stream) {
  const float* x     = (const float*)d_in[0];
  const float* w_dw1 = (const float*)d_in[1];
  const float* b_dw1 = (const float*)d_in[2];
  const float* w_dw2 = (const float*)d_in[3];
  const float* b_dw2 = (const float*)d_in[4];
  const float* w_qk  = (const float*)d_in[5];
  const float* b_qk  = (const float*)d_in[6];

  float* out   = (float*)d_out;
  float* feats = out + (size_t)B_ * GC_ * HW_;          // after mem

  float* qkws = (float*)d_ws;                           // B*256*HW elu'd q||k
  float* kmws = qkws + (size_t)B_ * 2 * GC_ * HW_;      // B*4*32
  float* kvws = kmws + (size_t)B_ * HEADS_ * HD_;       // B*4*32*32

  // 1) mem passthrough
  copy_mem_kernel<<<(B_ * GC_ * HW_ / 4) / 256, 256, 0, stream>>>(x, out);
  // 2) depthwise 3x3 (in ch 256.., out feats ch 128..)
  dwconv_kernel<3, 1><<<B_ * GC_, 256, 0, stream>>>(x, w_dw1, b_dw1, feats, 256, 128);
  // 3) depthwise 7x7 (in ch 384.., out feats ch 256..)
  dwconv_kernel<7, 3><<<B_ * GC_, 256, 0, stream>>>(x, w_dw2, b_dw2, feats, 384, 256);
  // 4) qk GEMM + bias + elu+1 (WMMA f32, async-to-LDS staging)
  qk_gemm_kernel<<<B_ * 49, 256, 0, stream>>>(x, w_qk, b_qk, qkws);
  // 5) kmean
  kmean_kernel<<<B_ * HEADS_, 256, 0, stream>>>(qkws, kmws);
  // 6) kv = k^T v / HW
  kv_kernel<<<B_ * HEADS_, 256, 0, stream>>>(qkws, x, kvws);
  // 7) att = (q kv) * z  (WMMA f32)
  dim3 g(49, B_ * HEADS_);
  att_kernel<<<g, 256, 0, stream>>>(qkws, kvws, kmws, feats);
}